// DCNv2_85083302134365
// MI455X (gfx1250) — compile-verified
//
#include <hip/hip_runtime.h>
#include <math.h>

typedef __attribute__((ext_vector_type(2))) float v2f;
typedef __attribute__((ext_vector_type(8))) float v8f;

#define BATCH 8
#define CIN   64
#define Hdim  128
#define Wdim  128
#define K2    9
#define CK    576      // CIN * K2
#define OCH   64
#define OFFCH 18
#define NCH   27       // 18 offset channels + 9 mask channels
#define PLANE (Hdim * Wdim)

static __device__ __forceinline__ int iclamp(int v, int lo, int hi) {
    return v < lo ? lo : (v > hi ? hi : v);
}

// ---------------------------------------------------------------------------
// Kernel 0: one-time reorder of w_dcn (O, c*9+k) -> (O, k*64+c) into workspace
// ---------------------------------------------------------------------------
__global__ __launch_bounds__(256)
void dcn_reorder_wdcn(const float* __restrict__ w_dcn, float* __restrict__ w_r)
{
    const int i = blockIdx.x * 256 + threadIdx.x;
    if (i < OCH * CK) {
        const int o = i / CK;
        const int r = i - o * CK;      // dst reduce index = k*64 + c
        const int k = r >> 6;
        const int c = r & 63;
        w_r[i] = w_dcn[o * CK + c * K2 + k];
    }
}

// ---------------------------------------------------------------------------
// Kernel 1: offset + mask 3x3 conv as implicit GEMM on V_WMMA_F32_16X16X4_F32
//   Reduce order r = t*64 + c: tap math + bounds check hoisted out of the
//   channel loop -> branch-free inner loop (2 loads, 2 ds_load_b64, 2 WMMA).
// ---------------------------------------------------------------------------
__global__ __launch_bounds__(256)
void dcn_offmask_kernel(const float* __restrict__ x,
                        const float* __restrict__ w_off,
                        const float* __restrict__ b_off,
                        const float* __restrict__ w_mask,
                        const float* __restrict__ b_mask,
                        float* __restrict__ off_buf,    // (B,18,H,W)
                        float* __restrict__ mask_buf)   // (B,9,H,W)
{
    constexpr int WSTR = CK + 2;        // 578 dwords: 578 % 64 = 2 -> lanes on
    __shared__ float Wlds[NCH * WSTR];  // distinct banks; 62424 B static LDS

    const int tid = threadIdx.x;
    const int b   = blockIdx.x >> 7;
    const int h   = blockIdx.x & 127;

    // Stage combined (27 x 576) weight matrix, permuted (c*9+t) -> (t*64+c).
    for (int i = tid; i < NCH * CK; i += 256) {
        const int n   = i / CK;
        const int rem = i - n * CK;
        const int t   = rem >> 6;
        const int c   = rem & 63;
        const float* src = (n < OFFCH) ? (w_off + (size_t)n * CK)
                                       : (w_mask + (size_t)(n - OFFCH) * CK);
        Wlds[n * WSTR + rem] = src[c * K2 + t];
    }
    __syncthreads();

    const int wave  = tid >> 5;
    const int lane  = tid & 31;
    const int w0    = wave << 4;          // 8 waves cover one 128-wide row
    const int mlane = lane & 15;          // A-fragment M index
    const int khalf = (lane >> 4) << 1;   // 0 or 2: K pair held by this lane

    const float* xb = x + (size_t)b * CIN * PLANE;

    v8f acc0 = {0.f,0.f,0.f,0.f,0.f,0.f,0.f,0.f};
    v8f acc1 = {0.f,0.f,0.f,0.f,0.f,0.f,0.f,0.f};

    const int n0 = mlane;                 // N-tile 0 (always offset channels)
    const int n1 = mlane + 16;            // N-tile 1 (valid when < 27)
    const float* wrow0 = &Wlds[n0 * WSTR];
    const float* wrow1 = &Wlds[(n1 < NCH ? n1 : NCH - 1) * WSTR];

    #pragma unroll
    for (int t = 0; t < K2; ++t) {
        const int ki = t / 3;             // compile-time constants (unrolled)
        const int kj = t - ki * 3;
        const int y  = h + ki - 1;
        const int xx = w0 + mlane + kj - 1;
        const float vmul =
            ((unsigned)y < (unsigned)Hdim && (unsigned)xx < (unsigned)Wdim) ? 1.f : 0.f;
        const float* xp = xb + (size_t)(iclamp(y, 0, Hdim - 1) * Wdim
                                        + iclamp(xx, 0, Wdim - 1));
        const float* wr0 = wrow0 + t * CIN;
        const float* wr1 = wrow1 + t * CIN;

        #pragma unroll 2
        for (int cc = 0; cc < CIN; cc += 4) {
            const int cA = cc + khalf;
            v2f a;
            a[0] = xp[(size_t)cA * PLANE] * vmul;        // branch-free, coalesced
            a[1] = xp[(size_t)(cA + 1) * PLANE] * vmul;
            const v2f b0 = *(const v2f*)&wr0[cA];        // ds_load_b64
            const v2f b1 = *(const v2f*)&wr1[cA];
            acc0 = __builtin_amdgcn_wmma_f32_16x16x4_f32(false, a, false, b0,
                                                         (short)0, acc0, false, false);
            acc1 = __builtin_amdgcn_wmma_f32_16x16x4_f32(false, a, false, b1,
                                                         (short)0, acc1, false, false);
        }
    }

    // Epilogue: C layout VGPR r -> M = r (lanes 0-15) / r+8 (lanes 16-31)
    const int mrow = (lane >> 4) << 3;
    #pragma unroll
    for (int r = 0; r < 8; ++r) {
        const int wp = w0 + r + mrow;
        off_buf[(((size_t)b * OFFCH + n0) * Hdim + h) * Wdim + wp] = acc0[r] + b_off[n0];
        if (n1 < NCH) {
            if (n1 < OFFCH) {
                off_buf[(((size_t)b * OFFCH + n1) * Hdim + h) * Wdim + wp] =
                    acc1[r] + b_off[n1];
            } else {
                const int mch = n1 - OFFCH;
                const float v = acc1[r] + b_mask[mch];
                mask_buf[(((size_t)b * K2 + mch) * Hdim + h) * Wdim + wp] =
                    1.f / (1.f + __expf(-v));
            }
        }
    }
}

// ---------------------------------------------------------------------------
// Kernel 2: fused deformable sampling + output GEMM (reduce order r = k*64+c)
// ---------------------------------------------------------------------------
__global__ __launch_bounds__(128)
void dcn_sample_gemm_kernel(const float* __restrict__ x,
                            const float* __restrict__ off_buf,
                            const float* __restrict__ mask_buf,
                            const float* __restrict__ w_r,   // (O, k*64+c) reordered
                            const float* __restrict__ b_dcn,
                            float* __restrict__ out)
{
    constexpr int MT   = 16;
    constexpr int SSTR = CK + 2;            // 578: conflict-free A-fragment rows
    __shared__ float S[MT * SSTR];          // sampled A-panel, 36992 B
    __shared__ float CW[MT * K2 * 4];       // mask-folded bilinear corner weights
    __shared__ int   CI[MT * K2 * 4];       // clamped corner coords y0,y1,x0,x1

    const int tid = threadIdx.x;
    const int t   = blockIdx.x;
    const int w0  = (t & 7) << 4;
    const int h   = (t >> 3) & 127;
    const int b   = t >> 10;

    const float* xb = x + (size_t)b * CIN * PLANE;

    // Phase 1a: per-(m,k) corner data -- shared across all 64 input channels
    for (int e = tid; e < MT * K2; e += 128) {
        const int m  = e / K2;
        const int k  = e - m * K2;
        const int wp = w0 + m;
        const float dy = off_buf[(((size_t)b * OFFCH + 2 * k    ) * Hdim + h) * Wdim + wp];
        const float dx = off_buf[(((size_t)b * OFFCH + 2 * k + 1) * Hdim + h) * Wdim + wp];
        const float mk = mask_buf[(((size_t)b * K2 + k) * Hdim + h) * Wdim + wp];
        const int ki = k / 3;
        const int kj = k - ki * 3;
        const float py  = (float)(h + ki - 1) + dy;
        const float px  = (float)(wp + kj - 1) + dx;
        const float y0f = floorf(py), x0f = floorf(px);
        const float wy1 = py - y0f,  wy0 = 1.f - wy1;
        const float wx1 = px - x0f,  wx0 = 1.f - wx1;
        const int y0 = (int)y0f, x0i = (int)x0f;
        const int y1 = y0 + 1,   x1i = x0i + 1;
        const float fy0 = ((unsigned)y0  < (unsigned)Hdim) ? 1.f : 0.f;
        const float fy1 = ((unsigned)y1  < (unsigned)Hdim) ? 1.f : 0.f;
        const float fx0 = ((unsigned)x0i < (unsigned)Wdim) ? 1.f : 0.f;
        const float fx1 = ((unsigned)x1i < (unsigned)Wdim) ? 1.f : 0.f;
        CW[e * 4 + 0] = wy0 * wx0 * mk * fy0 * fx0;
        CW[e * 4 + 1] = wy0 * wx1 * mk * fy0 * fx1;
        CW[e * 4 + 2] = wy1 * wx0 * mk * fy1 * fx0;
        CW[e * 4 + 3] = wy1 * wx1 * mk * fy1 * fx1;
        CI[e * 4 + 0] = iclamp(y0,  0, Hdim - 1);
        CI[e * 4 + 1] = iclamp(y1,  0, Hdim - 1);
        CI[e * 4 + 2] = iclamp(x0i, 0, Wdim - 1);
        CI[e * 4 + 3] = iclamp(x1i, 0, Wdim - 1);
    }
    __syncthreads();

    // Phase 1b: materialize sampled panel S[m][k*64+c] (k,c via shifts only)
    for (int e = tid; e < MT * CK; e += 128) {
        const int m  = e / CK;
        const int r  = e - m * CK;      // r = k*64 + c
        const int k  = r >> 6;
        const int c  = r & 63;
        const int ce = (m * K2 + k) * 4;
        const float* xc = xb + (size_t)c * PLANE;
        const int y0 = CI[ce + 0], y1 = CI[ce + 1];
        const int x0 = CI[ce + 2], x1 = CI[ce + 3];
        S[m * SSTR + r] = CW[ce + 0] * xc[y0 * Wdim + x0]
                        + CW[ce + 1] * xc[y0 * Wdim + x1]
                        + CW[ce + 2] * xc[y1 * Wdim + x0]
                        + CW[ce + 3] * xc[y1 * Wdim + x1];
    }
    __syncthreads();

    // Phase 2: GEMM, one 16-wide N-tile of O per wave
    const int wave  = tid >> 5;
    const int lane  = tid & 31;
    const int nb    = wave << 4;
    const int nl    = lane & 15;
    const int khalf = (lane >> 4) << 1;
    const float* srow = &S[(lane & 15) * SSTR];
    const float* brow = &w_r[(size_t)(nb + nl) * CK];

    v8f acc = {0.f,0.f,0.f,0.f,0.f,0.f,0.f,0.f};
    #pragma unroll 4
    for (int kk = 0; kk < CK; kk += 4) {
        const v2f a  = *(const v2f*)&srow[kk + khalf];   // ds_load_b64
        const v2f bb = *(const v2f*)&brow[kk + khalf];   // global_load_b64
        acc = __builtin_amdgcn_wmma_f32_16x16x4_f32(false, a, false, bb,
                                                    (short)0, acc, false, false);
    }

    const int o    = nb + nl;
    const float bz = b_dcn[o];
    const int mrow = (lane >> 4) << 3;
    float* orow = out + (((size_t)b * OCH + o) * Hdim + h) * Wdim + w0;
    #pragma unroll
    for (int r = 0; r < 8; ++r)
        orow[r + mrow] = acc[r] + bz;
}

// ---------------------------------------------------------------------------
extern "C" void kernel_launch(void* const* d_in, const int* in_sizes, int n_in,
                              void* d_out, int out_size, void* d_ws, size_t ws_size,
                              hipStream_t stream)
{
    const float* x      = (const float*)d_in[0];
    const float* w_off  = (const float*)d_in[1];
    const float* b_off  = (const float*)d_in[2];
    const float* w_mask = (const float*)d_in[3];
    const float* b_mask = (const float*)d_in[4];
    const float* w_dcn  = (const float*)d_in[5];
    const float* b_dcn  = (const float*)d_in[6];
    float* out = (float*)d_out;

    // Workspace: off (B*18*H*W) + mask (B*9*H*W) + reordered w_dcn (O*CK)
    float* off_buf  = (float*)d_ws;
    float* mask_buf = off_buf  + (size_t)BATCH * OFFCH * PLANE;
    float* w_r      = mask_buf + (size_t)BATCH * K2 * PLANE;

    dcn_reorder_wdcn<<<(OCH * CK + 255) / 256, 256, 0, stream>>>(w_dcn, w_r);

    dcn_offmask_kernel<<<BATCH * Hdim, 256, 0, stream>>>(
        x, w_off, b_off, w_mask, b_mask, off_buf, mask_buf);

    dcn_sample_gemm_kernel<<<BATCH * Hdim * (Wdim / 16), 128, 0, stream>>>(
        x, off_buf, mask_buf, w_r, b_dcn, out);
}